// PolymerDistance_38826504356268
// MI455X (gfx1250) — compile-verified
//
#include <hip/hip_runtime.h>
#include <math.h>

// ---------------------------------------------------------------------------
// PolymerDistance (Kabsch-style per-molecule loss) for MI455X / gfx1250.
//
// Bandwidth-bound segmented reduction. The 21 per-molecule accumulators
// (3x3 cross sums, coord sums, square sums) are fused into ONE
// V_WMMA_F32_16X16X4_F32 accumulator per wave using augmented fragments:
//   A (16x4, M=rows): rows 0-2 = coords2 (y), row 3 = 1, rows 4-6 = coords1 (x)
//   B (4x16, N=cols): cols 0-2 = coords1 (x), col 3 = 1, cols 4-6 = coords2 (y)
//   D[j][n] (j,n<3)   = sum y_j x_n      (cross-cov raw sums)
//   D[j][3] / D[3][n] = sum y_j / sum x_n
//   D[j][4+j], D[4+k][k] = sum y_j^2, sum x_k^2
// The constant rows (1.0) / dead rows (0.0) are materialized in small LDS
// regions so the inner loop is pure ds_load_2addr + v_wmma (no cndmask).
// Global->LDS staging uses GLOBAL_LOAD_ASYNC_TO_LDS_B128 (ASYNCcnt) with
// double-buffered tiles when the builtin is available.
// ---------------------------------------------------------------------------

typedef __attribute__((ext_vector_type(2))) float v2f;
typedef __attribute__((ext_vector_type(8))) float v8f;
typedef __attribute__((ext_vector_type(4))) int   v4i;

#define THREADS     256
#define NWAVES      (THREADS / 32)
#define TILE_ATOMS  512
#define WAVE_ATOMS  (TILE_ATOMS / NWAVES)      /* 64 atoms per wave per tile */
#define STAGE_F     (TILE_ATOMS * 3)           /* floats per array per tile  */

/* unified LDS layout (floats) */
#define BUF_STRIDE  (2 * STAGE_F)              /* X + Y per buffer = 3072    */
#define ONES_OFF    (2 * BUF_STRIDE)           /* 6144: 192 floats of 1.0    */
#define ZEROS_OFF   (ONES_OFF + 192)           /* 6336: 192 floats of 0.0    */
#define RED_OFF     (ZEROS_OFF + 192)          /* 6528: NWAVES*64            */
#define FIN_OFF     (RED_OFF + NWAVES * 64)    /* 7040: 64                   */
#define LDS_FLOATS  (FIN_OFF + 64)             /* 7104 floats = 28416 B      */

#if defined(__AMDGCN__) && \
    __has_builtin(__builtin_amdgcn_global_load_async_to_lds_b128) && \
    __has_builtin(__builtin_amdgcn_s_wait_asynccnt)
#define USE_ASYNC 1
#else
#define USE_ASYNC 0
#endif

#if USE_ASYNC
/* pointer-to-global-v4i and pointer-to-LDS-v4i, per the builtin's prototype
   (param printed by clang as 'int __vector(4) __device__ *'). */
typedef __attribute__((address_space(1))) v4i* gptr_b128;
typedef __attribute__((address_space(3))) v4i* lptr_b128;

__device__ __forceinline__ void async_copy16(const float* g, float* l) {
    __builtin_amdgcn_global_load_async_to_lds_b128(
        (gptr_b128)(g), (lptr_b128)(l), 0, 0);
}
#endif

__device__ __forceinline__ void stage_tile(const float* __restrict__ g1,
                                           const float* __restrict__ g2,
                                           float* __restrict__ sX,
                                           float* __restrict__ sY,
                                           int nF, int tid)
{
    const int nV4 = nF >> 2;
#if USE_ASYNC
    for (int i = tid; i < nV4; i += THREADS) {
        async_copy16(g1 + 4 * i, sX + 4 * i);
        async_copy16(g2 + 4 * i, sY + 4 * i);
    }
#else
    for (int i = tid; i < nV4; i += THREADS) {
        ((float4*)sX)[i] = ((const float4*)g1)[i];
        ((float4*)sY)[i] = ((const float4*)g2)[i];
    }
#endif
    for (int i = (nV4 << 2) + tid; i < nF; i += THREADS) {  /* tail floats */
        sX[i] = g1[i];
        sY[i] = g2[i];
    }
    for (int i = nF + tid; i < STAGE_F; i += THREADS) {     /* zero-pad */
        sX[i] = 0.f;
        sY[i] = 0.f;
    }
}

__global__ __launch_bounds__(THREADS)
void PolymerDistance_kernel(const float* __restrict__ coords1,
                            const float* __restrict__ coords2,
                            float* __restrict__ out,
                            int apm /* atoms per molecule */)
{
    __shared__ float lds[LDS_FLOATS];

    const int tid  = threadIdx.x;
    const int lane = tid & 31;
    const int wave = tid >> 5;
    const int mol  = blockIdx.x;
    const size_t molBase = (size_t)mol * (size_t)apm * 3u;

    // ---- per-lane fragment classification (wave32, 16x16x4 f32 layouts) ----
    // A 16x4 (32-bit): v0 <- K=0 (lanes 0-15) / K=2 (lanes 16-31)
    //                  v1 <- K=1 (lanes 0-15) / K=3 (lanes 16-31)
    // B 4x16 mirrors:  v0 <- row K=0 / K=2, v1 <- row K=1 / K=3, N = lane%16
    const int r    = lane & 15;
    const int half = lane >> 4;            // 0: K={0,1}, 1: K={2,3}

    // class: 0 = first data region, 1 = ones, 2 = second data region, 3 = zeros
    const int  cls  = (r < 3) ? 0 : (r == 3) ? 1 : (r < 7) ? 2 : 3;
    const int  comp = (cls == 0) ? r : (cls == 2) ? (r - 4) : 0;
    const bool data = (cls == 0) || (cls == 2);
    // A rows 0-2 <- Y (coords2, at +STAGE_F), rows 4-6 <- X (coords1, at +0)
    const int offA = (cls == 0) ? STAGE_F : 0;
    // B cols 0-2 <- X, cols 4-6 <- Y
    const int offB = (cls == 2) ? STAGE_F : 0;
    const int constRegion = (cls == 1) ? ONES_OFF : ZEROS_OFF;

    // per-lane base index (buffer term added per tile for data lanes only)
    const int baseA = data ? (offA + wave * (WAVE_ATOMS * 3) + 6 * half + comp)
                           : (constRegion + 6 * half);
    const int baseB = data ? (offB + wave * (WAVE_ATOMS * 3) + 6 * half + comp)
                           : (constRegion + 6 * half);
    const int bufMul = data ? BUF_STRIDE : 0;

    // ---- fill the ones/zeros constant regions (covers the +12/iter sweep) --
    for (int i = tid; i < 192; i += THREADS) {
        lds[ONES_OFF  + i] = 1.0f;
        lds[ZEROS_OFF + i] = 0.0f;
    }

    v8f acc0 = {0.f, 0.f, 0.f, 0.f, 0.f, 0.f, 0.f, 0.f};
    v8f acc1 = {0.f, 0.f, 0.f, 0.f, 0.f, 0.f, 0.f, 0.f};

    const int tiles = (apm + TILE_ATOMS - 1) / TILE_ATOMS;

    // prologue: stage tile 0 into buffer 0
    {
        const int nAtoms = (apm < TILE_ATOMS) ? apm : TILE_ATOMS;
        stage_tile(coords1 + molBase, coords2 + molBase,
                   lds, lds + STAGE_F, nAtoms * 3, tid);
    }

    for (int t = 0; t < tiles; ++t) {
        const int buf = t & 1;
#if USE_ASYNC
        __builtin_amdgcn_s_wait_asynccnt(0);
#endif
        __syncthreads();

        // issue next tile into the other buffer (overlaps with compute)
        if (t + 1 < tiles) {
            const int tBase  = (t + 1) * TILE_ATOMS;
            const int nAtoms = (apm - tBase < TILE_ATOMS) ? (apm - tBase)
                                                          : TILE_ATOMS;
            const size_t g = molBase + (size_t)tBase * 3u;
            float* s = lds + (buf ^ 1) * BUF_STRIDE;
            stage_tile(coords1 + g, coords2 + g, s, s + STAGE_F,
                       nAtoms * 3, tid);
        }

        // ---- per-wave WMMA accumulation: WAVE_ATOMS atoms, 4 per WMMA ----
        int idxA = baseA + (buf ? bufMul : 0);
        int idxB = baseB + (buf ? bufMul : 0);
        #pragma unroll
        for (int it = 0; it < WAVE_ATOMS / 4; it += 2) {
            {
                v2f A = { lds[idxA], lds[idxA + 3] };
                v2f B = { lds[idxB], lds[idxB + 3] };
                acc0 = __builtin_amdgcn_wmma_f32_16x16x4_f32(
                    false, A, false, B, (short)0, acc0, false, false);
                idxA += 12; idxB += 12;
            }
            {
                v2f A = { lds[idxA], lds[idxA + 3] };
                v2f B = { lds[idxB], lds[idxB + 3] };
                acc1 = __builtin_amdgcn_wmma_f32_16x16x4_f32(
                    false, A, false, B, (short)0, acc1, false, false);
                idxA += 12; idxB += 12;
            }
        }
    }

    // ---- cross-wave reduction of the 7x7 useful block of D ----
    // C/D layout: VGPR j, lanes 0-15 hold D[M=j][N=lane]. We need M,N in 0..6.
    float* red = lds + RED_OFF;
    float* fin = lds + FIN_OFF;
    __syncthreads();                    // staging region reuse-safe barrier
    if (lane < 8) {
        #pragma unroll
        for (int j = 0; j < 7; ++j)
            red[wave * 64 + j * 8 + lane] = acc0[j] + acc1[j];
    }
    __syncthreads();
    if (tid < 56) {
        float v = 0.f;
        #pragma unroll
        for (int w = 0; w < NWAVES; ++w) v += red[w * 64 + tid];
        fin[tid] = v;
    }
    __syncthreads();

    // ---- one lane: 3x3 closed-form eigensolve + output ----
    if (tid == 0) {
        const float invN = 1.0f / (float)apm;
        float m1[3], m2[3];
        #pragma unroll
        for (int k = 0; k < 3; ++k) m1[k] = fin[24 + k] * invN;       // D[3][k]
        #pragma unroll
        for (int j = 0; j < 3; ++j) m2[j] = fin[j * 8 + 3] * invN;    // D[j][3]

        float cov[3][3];
        #pragma unroll
        for (int j = 0; j < 3; ++j)
            #pragma unroll
            for (int k = 0; k < 3; ++k)
                cov[j][k] = fin[j * 8 + k] * invN - m2[j] * m1[k];

        float var1 = 0.f, var2 = 0.f;
        #pragma unroll
        for (int k = 0; k < 3; ++k) {
            var1 += fin[(4 + k) * 8 + k] * invN - m1[k] * m1[k];      // sum x^2
            var2 += fin[k * 8 + 4 + k] * invN - m2[k] * m2[k];        // sum y^2
        }
        var1 *= (1.0f / 3.0f);
        var2 *= (1.0f / 3.0f);

        const float det =
            cov[0][0] * (cov[1][1] * cov[2][2] - cov[1][2] * cov[2][1]) -
            cov[0][1] * (cov[1][0] * cov[2][2] - cov[1][2] * cov[2][0]) +
            cov[0][2] * (cov[1][0] * cov[2][1] - cov[1][1] * cov[2][0]);

        // B = cov^T * cov (symmetric PSD); sigma_i = sqrt(eig_i(B))
        float b00 = 0.f, b01 = 0.f, b02 = 0.f, b11 = 0.f, b12 = 0.f, b22 = 0.f;
        #pragma unroll
        for (int k = 0; k < 3; ++k) {
            b00 += cov[k][0] * cov[k][0];
            b01 += cov[k][0] * cov[k][1];
            b02 += cov[k][0] * cov[k][2];
            b11 += cov[k][1] * cov[k][1];
            b12 += cov[k][1] * cov[k][2];
            b22 += cov[k][2] * cov[k][2];
        }
        const float q  = (b00 + b11 + b22) * (1.0f / 3.0f);
        const float p1 = b01 * b01 + b02 * b02 + b12 * b12;
        const float d0 = b00 - q, d1 = b11 - q, d2 = b22 - q;
        const float p2 = d0 * d0 + d1 * d1 + d2 * d2 + 2.0f * p1;
        float e1, e2, e3;
        if (p2 <= 1e-30f) {
            e1 = e2 = e3 = q;
        } else {
            const float p  = sqrtf(p2 * (1.0f / 6.0f));
            const float ip = 1.0f / p;
            const float c00 = d0 * ip, c11 = d1 * ip, c22 = d2 * ip;
            const float c01 = b01 * ip, c02 = b02 * ip, c12 = b12 * ip;
            float rr = 0.5f * (c00 * (c11 * c22 - c12 * c12) -
                               c01 * (c01 * c22 - c12 * c02) +
                               c02 * (c01 * c12 - c11 * c02));
            rr = fminf(1.0f, fmaxf(-1.0f, rr));
            const float phi = acosf(rr) * (1.0f / 3.0f);
            e1 = q + 2.0f * p * cosf(phi);
            e3 = q + 2.0f * p * cosf(phi + 2.0943951023931953f); // + 2*pi/3
            e2 = 3.0f * q - e1 - e3;                             // e1>=e2>=e3
        }
        const float s1 = sqrtf(fmaxf(e1, 0.f));
        const float s2 = sqrtf(fmaxf(e2, 0.f));
        const float s3 = sqrtf(fmaxf(e3, 0.f));   // smallest singular value
        const float sgn = (det < 0.f) ? -1.f : 1.f;
        const float sigma_mean = (s1 + s2 + sgn * s3) * (1.0f / 3.0f);
        out[mol] = var1 + var2 - 2.0f * sigma_mean;
    }
}

extern "C" void kernel_launch(void* const* d_in, const int* in_sizes, int n_in,
                              void* d_out, int out_size, void* d_ws, size_t ws_size,
                              hipStream_t stream) {
    (void)d_ws; (void)ws_size; (void)n_in;
    const float* coords1 = (const float*)d_in[0];
    const float* coords2 = (const float*)d_in[1];
    float* out = (float*)d_out;

    const int n_mol   = out_size;                       // one output per molecule
    if (n_mol <= 0) return;
    const int n_atoms = in_sizes[0] / 3;
    const int apm     = n_atoms / n_mol;                // contiguous equal segments

    PolymerDistance_kernel<<<n_mol, THREADS, 0, stream>>>(coords1, coords2, out, apm);
}